// Decoding_13460427506032
// MI455X (gfx1250) — compile-verified
//
#include <hip/hip_runtime.h>
#include <hip/hip_bf16.h>

#define NG 5000
#define NL 25          // latent (== reflatent rows R)
#define NR 25
#define NC 128         // vertex heights
#define NK 127         // bins

typedef __attribute__((ext_vector_type(2))) float v2f;
typedef __attribute__((ext_vector_type(8))) float v8f;

// -------- Phase 1: per-gene fused kernel -----------------------------------
// widths softmax + cumwidths, delta = reflatent @ W[g] via V_WMMA_F32_16X16X4_F32,
// heights = exp(uh+delta)/trapezoid_area, and sum(W^2) partial for the KL term.
__global__ __launch_bounds__(512) void gene_kernel(
    const float* __restrict__ refl,      // [25,25]
    const float* __restrict__ W,         // [NG,25,128]
    const float* __restrict__ uh,        // [NG,128]
    const float* __restrict__ uw,        // [NG,127]
    float* __restrict__ heights,         // [NR,NG,NC]
    float* __restrict__ widths,          // [NG,NK]
    float* __restrict__ cumw,            // [NG,NC]
    float* __restrict__ sq_part)         // [NG]
{
  __shared__ float s_h[32 * NC];         // 16 KB: rows 25..31 are scratch pad
  __shared__ float s_w[NC];              // widths after softmax
  __shared__ float s_cw[NC];             // cumwidths
  __shared__ float s_area[32];
  __shared__ float s_red[16];

  const int g    = blockIdx.x;
  const int tid  = threadIdx.x;
  const int lane = tid & 31;
  const int wave = tid >> 5;

  // ---- widths softmax (wave 0) ----
  if (tid < NK) s_w[tid] = uw[(size_t)g * NK + tid];
  __syncthreads();
  if (wave == 0) {
    float m = -3.4e38f;
    for (int k = lane; k < NK; k += 32) m = fmaxf(m, s_w[k]);
    for (int o = 16; o; o >>= 1) m = fmaxf(m, __shfl_xor(m, o, 32));
    float s = 0.f;
    for (int k = lane; k < NK; k += 32) s += __expf(s_w[k] - m);
    for (int o = 16; o; o >>= 1) s += __shfl_xor(s, o, 32);
    float inv = 1.f / s;
    for (int k = lane; k < NK; k += 32) s_w[k] = __expf(s_w[k] - m) * inv;
  }
  __syncthreads();
  if (tid == 0) {
    float c = 0.f;
    s_cw[0] = 0.f;
    for (int k = 0; k < NK - 1; ++k) { c += s_w[k]; s_cw[k + 1] = c; }
    s_cw[NK] = 1.0f;                     // reference forces last cumwidth to 1
  }

  // ---- delta = reflatent(25x25) @ W[g](25x128) via WMMA f32 16x16x4 ----
  // wave -> (m_tile, n_tile); 2 x 8 tiles cover 32x128 (rows >=25 padded).
  const int m_tile = wave >> 3;
  const int n_tile = wave & 7;
  const int half   = lane >> 4;          // 0 or 1
  const int l16    = lane & 15;
  const int mrow   = m_tile * 16 + l16;  // A-matrix row (M) for this lane
  const int ncol   = n_tile * 16 + l16;  // B/D column (N) for this lane
  const float* Wg  = W + (size_t)g * NL * NC;

  v8f acc = {0.f, 0.f, 0.f, 0.f, 0.f, 0.f, 0.f, 0.f};
  float ksum = 0.f;                      // sum of W^2 (m_tile==0 waves only)
  const int mr_c = min(mrow, NL - 1);

#pragma unroll
  for (int k0 = 0; k0 < 28; k0 += 4) {   // K=25 padded to 28
    const int ka = k0 + 2 * half;        // K index held in VGPR v=0
    const int kb = ka + 1;               // K index held in VGPR v=1
    // branchless (cndmask) zero-padding: EXEC stays all-ones for WMMA
    float ax = refl[mr_c * NL + min(ka, NL - 1)];
    float ay = refl[mr_c * NL + min(kb, NL - 1)];
    ax = (mrow < NR && ka < NL) ? ax : 0.f;
    ay = (mrow < NR && kb < NL) ? ay : 0.f;
    float bx = Wg[(size_t)min(ka, NL - 1) * NC + ncol];
    float by = Wg[(size_t)min(kb, NL - 1) * NC + ncol];
    bx = (ka < NL) ? bx : 0.f;
    by = (kb < NL) ? by : 0.f;
    if (m_tile == 0) ksum += bx * bx + by * by;   // each W elem counted once
    v2f a; a.x = ax; a.y = ay;
    v2f b; b.x = bx; b.y = by;
    // (neg_a, A, neg_b, B, c_mod, C, reuse_a, reuse_b)
    acc = __builtin_amdgcn_wmma_f32_16x16x4_f32(
        false, a, false, b, (short)0, acc, false, false);
  }

  // ---- h_exp = exp(uh + delta) into LDS (D layout: m = 8*half + v) ----
  // Unconditional stores: rows 25..31 land in the scratch pad region, so no
  // divergent exec-mask branches between the WMMAs and the stores.
  const float uhv = uh[(size_t)g * NC + ncol];
#pragma unroll
  for (int v = 0; v < 8; ++v) {
    const int m = m_tile * 16 + 8 * half + v;     // 0..31
    s_h[m * NC + ncol] = __expf(uhv + acc[v]);
  }

  // ---- KL sum(W^2) block partial (deterministic, no atomics) ----
  for (int o = 16; o; o >>= 1) ksum += __shfl_xor(ksum, o, 32);
  if (lane == 0) s_red[wave] = ksum;
  __syncthreads();                       // covers s_h, s_red, s_cw
  if (tid == 0) {
    float t = 0.f;
    for (int i = 0; i < 16; ++i) t += s_red[i];
    sq_part[g] = t;
  }

  // ---- trapezoid areas per r, then normalized heights to global ----
  for (int r = wave; r < NR; r += 16) {
    float p = 0.f;
    for (int k = lane; k < NK; k += 32)
      p += 0.5f * (s_h[r * NC + k] + s_h[r * NC + k + 1]) * s_w[k];
    for (int o = 16; o; o >>= 1) p += __shfl_xor(p, o, 32);
    if (lane == 0) s_area[r] = p;
  }
  __syncthreads();
  for (int idx = tid; idx < NR * NC; idx += 512) {
    const int r = idx >> 7, c = idx & (NC - 1);
    heights[((size_t)r * NG + g) * NC + c] = s_h[idx] / s_area[r];
  }
  if (tid < NK) widths[(size_t)g * NK + tid] = s_w[tid];
  if (tid < NC) cumw[(size_t)g * NC + tid] = s_cw[tid];
}

// -------- Phase 2: per-cut spline log-prob ---------------------------------
__global__ __launch_bounds__(256) void cut_kernel(
    const float* __restrict__ x_all, const int* __restrict__ gix,
    const int* __restrict__ rix, const float* __restrict__ heights,
    const float* __restrict__ widths, const float* __restrict__ cumw,
    float* __restrict__ lik_part, int n)
{
  const int i = blockIdx.x * 256 + threadIdx.x;
  float v = 0.f;
  if (i < n) {
    const float x = x_all[i];
    const int g = gix[i];
    const int r = rix[i];
    const float* cw   = cumw + (size_t)g * NC;
    const float* hrow = heights + ((size_t)r * NG + g) * NC;
    __builtin_prefetch(hrow, 0, 0);      // global_prefetch_b8: warm L2 line
    // searchsorted over cw[1..126]: first idx with cw[idx] > x
    int lo = 1, hi = NK;
    while (lo < hi) {
      const int mid = (lo + hi) >> 1;
      const bool ge = (x >= cw[mid]);
      lo = ge ? mid + 1 : lo;
      hi = ge ? hi : mid;
    }
    const int b = lo - 1;                // in [0, 126] == [0, K-1]
    const float w_b  = widths[(size_t)g * NK + b];
    const float cw_l = cw[b];
    const float h_l  = hrow[b];
    const float h_r  = hrow[b + 1];
    const float alpha = (x - cw_l) / w_b;
    v = __logf(fmaf(alpha, h_r - h_l, h_l));
  }
  for (int o = 16; o; o >>= 1) v += __shfl_xor(v, o, 32);
  __shared__ float s[8];
  if ((threadIdx.x & 31) == 0) s[threadIdx.x >> 5] = v;
  __syncthreads();
  if (threadIdx.x == 0) {
    float t = 0.f;
    for (int j = 0; j < 8; ++j) t += s[j];
    lik_part[blockIdx.x] = t;
  }
}

// -------- Phase 3: deterministic final reduction ---------------------------
__global__ __launch_bounds__(256) void finalize_kernel(
    const float* __restrict__ sq_part, int nsq,
    const float* __restrict__ lik_part, int nlik,
    const int* __restrict__ n_cells, float* __restrict__ out)
{
  const int tid = threadIdx.x;
  float a = 0.f, b = 0.f;
  for (int i = tid; i < nsq;  i += 256) a += sq_part[i];
  for (int i = tid; i < nlik; i += 256) b += lik_part[i];
  for (int o = 16; o; o >>= 1) { a += __shfl_xor(a, o, 32); b += __shfl_xor(b, o, 32); }
  __shared__ float sa[8], sb[8];
  if ((tid & 31) == 0) { sa[tid >> 5] = a; sb[tid >> 5] = b; }
  __syncthreads();
  if (tid == 0) {
    float sumsq = 0.f, lik = 0.f;
    for (int j = 0; j < 8; ++j) { sumsq += sa[j]; lik += sb[j]; }
    const float NW = (float)NG * NL * NC;                 // 16e6 elements
    const float kl = NW * (-0.5f * logf(2.f * 3.14159265358979f)) - 0.5f * sumsq;
    out[0] = (-lik - kl) / (float)n_cells[0];
  }
}

extern "C" void kernel_launch(void* const* d_in, const int* in_sizes, int n_in,
                              void* d_out, int out_size, void* d_ws, size_t ws_size,
                              hipStream_t stream) {
  const float* refl = (const float*)d_in[0];   // [25,25]
  const float* W    = (const float*)d_in[1];   // [5000,25,128]
  const float* uh   = (const float*)d_in[2];   // [5000,128]
  const float* uw   = (const float*)d_in[3];   // [5000,127]
  const float* x    = (const float*)d_in[4];   // [1e6]
  const int*   gix  = (const int*)d_in[5];
  const int*   rix  = (const int*)d_in[6];
  const int*   ncll = (const int*)d_in[7];
  const int ncuts = in_sizes[4];

  // workspace layout (floats)
  float* ws       = (float*)d_ws;
  float* heights  = ws;                                   // 16,000,000
  float* widths   = ws + (size_t)NR * NG * NC;            //    635,000
  float* cumw     = widths + (size_t)NG * NK;             //    640,000
  float* sq_part  = cumw + (size_t)NG * NC;               //      5,000
  float* lik_part = sq_part + NG;                         //   <= 4,096

  const int cut_blocks = (ncuts + 255) / 256;

  gene_kernel<<<NG, 512, 0, stream>>>(refl, W, uh, uw, heights, widths, cumw, sq_part);
  cut_kernel<<<cut_blocks, 256, 0, stream>>>(x, gix, rix, heights, widths, cumw,
                                             lik_part, ncuts);
  finalize_kernel<<<1, 256, 0, stream>>>(sq_part, NG, lik_part, cut_blocks,
                                         ncll, (float*)d_out);
}